// RedoxAttention_36000415875574
// MI455X (gfx1250) — compile-verified
//
#include <hip/hip_runtime.h>
#include <math.h>

typedef float v2f __attribute__((ext_vector_type(2)));
typedef float v8f __attribute__((ext_vector_type(8)));

#define NB 4096    // molecules
#define NATOM 128  // atoms per molecule
#define ND 128     // feature dim
#define WSTRIDE 132  // padded K-stride for WT rows (4-bank step -> conflict-free b64)

__global__ __launch_bounds__(256) void redox_attn_wmma_kernel(
    const int*   __restrict__ atom_class,  // [B,128]
    const float* __restrict__ atom_state,  // [B,128,128]
    const float* __restrict__ Ea,          // [100,2]
    const float* __restrict__ Er,          // [100,2]
    const float* __restrict__ Wa,          // [128,2]
    const float* __restrict__ ba,          // [2]
    const float* __restrict__ Wr,          // [128,2]
    const float* __restrict__ br,          // [2]
    float*       __restrict__ out)         // [B,2]
{
    // WT[n][k]: B-matrix transposed per-lane. Rows 0-1 = Wa cols, 2-3 = Wr cols,
    // 4-15 = zero padding so every lane loads unconditionally (no exec masking).
    __shared__ __align__(16) float WT[16 * WSTRIDE];
    __shared__ int   cls[NATOM];
    __shared__ float sbuf[2][NATOM];  // masked attention logits
    __shared__ float rbuf[2][NATOM];  // redox predictions
    __shared__ float red[2][NATOM];   // reduction scratch

    const int b    = blockIdx.x;
    const int t    = threadIdx.x;
    const int lane = t & 31;
    const int wv   = t >> 5;       // wave id 0..7 -> 16-atom tile
    const int half = lane >> 4;    // 0/1 (K-split half of the wave)
    const int l15  = lane & 15;

    // Stage transposed/padded weights + atom classes.
    for (int i = t; i < 16 * WSTRIDE; i += 256) {
        const int n = i / WSTRIDE;
        const int k = i - n * WSTRIDE;
        float v = 0.0f;
        if (n < 2 && k < ND)      v = Wa[k * 2 + n];
        else if (n < 4 && k < ND) v = Wr[k * 2 + (n - 2)];
        WT[i] = v;
    }
    if (t < NATOM) cls[t] = atom_class[b * NATOM + t];
    __syncthreads();

    // --- WMMA accumulation: D[16x16] = sum_k A[16x4] * B[4x16] ---
    // A layout (16x4 f32): lanes 0-15 row M=l15 hold K=k0,k0+1; lanes 16-31 hold K=k0+2,k0+3.
    // B layout (4x16 f32): lanes 0-15 col N=l15 hold K=k0,k0+1; lanes 16-31 hold K=k0+2,k0+3.
    const float* gS   = atom_state + (size_t)b * NATOM * ND;
    const float* aRow = gS + (wv * 16 + l15) * ND;          // atom row this lane supplies
    const float* bRow = &WT[l15 * WSTRIDE];                 // weight column this lane supplies
    v8f acc = {};
    #pragma unroll 8
    for (int k0 = 0; k0 < ND; k0 += 4) {
        const int kk = k0 + 2 * half;
        v2f a  = *(const v2f*)(aRow + kk);   // global_load_b64, 8B aligned
        v2f bb = *(const v2f*)(bRow + kk);   // ds_load_b64, conflict-free
        acc = __builtin_amdgcn_wmma_f32_16x16x4_f32(
            /*neg_a=*/false, a, /*neg_b=*/false, bb,
            /*c_mod=*/(short)0, acc, /*reuse_a=*/false, /*reuse_b=*/false);
    }

    // --- Scatter D tile: cols 0-1 -> attention logits, cols 2-3 -> redox preds ---
    // D layout: VGPR r, lanes 0-15: M = r, lanes 16-31: M = r + 8; N = l15.
    const float ba0 = ba[0], ba1 = ba[1];
    const float br0 = br[0], br1 = br[1];
    #pragma unroll
    for (int r = 0; r < 8; ++r) {
        const int m   = wv * 16 + half * 8 + r;
        const float v = acc[r];
        const int cl  = cls[m];
        if (l15 < 2) {
            float s = v + ((l15 == 0) ? ba0 : ba1) + Ea[cl * 2 + l15];
            if (cl == 0) s = -3.4028234663852886e38f;  // finfo(f32).min mask
            sbuf[l15][m] = s;
        } else if (l15 < 4) {
            const int j = l15 - 2;
            rbuf[j][m] = v + ((j == 0) ? br0 : br1) + Er[cl * 2 + j];
        }
    }
    __syncthreads();

    // --- Masked softmax over atoms + attention-weighted pooling ---
    // 256 threads: channel j = t>>7, atom m = t&127.
    const int j = t >> 7;
    const int m = t & 127;
    const float sv = sbuf[j][m];

    red[j][m] = sv;
    __syncthreads();
    for (int st = 64; st > 0; st >>= 1) {
        if (m < st) red[j][m] = fmaxf(red[j][m], red[j][m + st]);
        __syncthreads();
    }
    const float mx = red[j][0];
    __syncthreads();

    const float e = expf(sv - mx);  // masked atoms: exp(-huge) -> 0
    red[j][m] = e;
    __syncthreads();
    for (int st = 64; st > 0; st >>= 1) {
        if (m < st) red[j][m] += red[j][m + st];
        __syncthreads();
    }
    const float denom = red[j][0];
    __syncthreads();

    red[j][m] = e * rbuf[j][m];
    __syncthreads();
    for (int st = 64; st > 0; st >>= 1) {
        if (m < st) red[j][m] += red[j][m + st];
        __syncthreads();
    }
    if (m == 0) out[b * 2 + j] = red[j][0] / denom;
}

extern "C" void kernel_launch(void* const* d_in, const int* in_sizes, int n_in,
                              void* d_out, int out_size, void* d_ws, size_t ws_size,
                              hipStream_t stream) {
    const int*   atom_class = (const int*)  d_in[0];
    const float* atom_state = (const float*)d_in[1];
    const float* Ea         = (const float*)d_in[2];
    const float* Er         = (const float*)d_in[3];
    const float* Wa         = (const float*)d_in[4];
    const float* ba         = (const float*)d_in[5];
    const float* Wr         = (const float*)d_in[6];
    const float* br         = (const float*)d_in[7];
    float* out = (float*)d_out;

    (void)in_sizes; (void)n_in; (void)out_size; (void)d_ws; (void)ws_size;

    redox_attn_wmma_kernel<<<NB, 256, 0, stream>>>(
        atom_class, atom_state, Ea, Er, Wa, ba, Wr, br, out);
}